// NestedMLP_67937792688174
// MI455X (gfx1250) — compile-verified
//
#include <hip/hip_runtime.h>
#include <hip/hip_bf16.h>

typedef __attribute__((ext_vector_type(2))) float v2f;
typedef __attribute__((ext_vector_type(4))) float v4f;
typedef __attribute__((ext_vector_type(8))) float v8f;

#define NUM_EXPERTS 4
#define TOKENS      16384     // B*N = 4*4096
#define DIM         1024      // D (w1 row stride)
#define HID         4096      // H (w2 row stride)
#define OUTD        1024      // OUT
#define TILE_M      16
#define LDS_PAD     4         // dword pad per LDS h-row -> conflict-free b64 reads

// ---------------------------------------------------------------------------
// Kernel 1: zero the output tensor (padding columns must be 0) + expert counts
// ---------------------------------------------------------------------------
__global__ void zero_out_kernel(float* __restrict__ out, int n4, int* __restrict__ counts) {
    int i = blockIdx.x * blockDim.x + threadIdx.x;
    if (i < n4) {
        v4f z = {0.f, 0.f, 0.f, 0.f};
        ((v4f*)out)[i] = z;
    }
    if (blockIdx.x == 0 && threadIdx.x < NUM_EXPERTS) counts[threadIdx.x] = 0;
}

// ---------------------------------------------------------------------------
// Kernel 2: route tokens -> per-expert index lists (atomic scatter)
// ---------------------------------------------------------------------------
__global__ void route_kernel(const int* __restrict__ mask, int* __restrict__ counts,
                             int* __restrict__ lists) {
    int t = blockIdx.x * blockDim.x + threadIdx.x;
    if (t >= TOKENS) return;
    int e = mask[t] & (NUM_EXPERTS - 1);
    int pos = atomicAdd(&counts[e], 1);
    lists[e * TOKENS + pos] = t;
}

__device__ __forceinline__ float gelu_exact(float v) {
    return 0.5f * v * (1.0f + erff(v * 0.7071067811865476f));
}

// ---------------------------------------------------------------------------
// Kernel 3: fused nested-MLP for one expert (compile-time widths).
// One block = 16 gathered tokens. 8 wave32 per block.
// Pass1: h[16][D_HID] = gelu(x16 @ w1[:D_HID,:D_IN]^T + b1) -> LDS
// Pass2: y[16][D_OUT] = h @ w2[:D_OUT,:D_HID]^T + b2        -> scattered store
// Each wave register-blocks TWO 16-wide N tiles (t0, t0+8) sharing one A
// fragment per K step: 2 WMMAs per A load.
// WMMA fragment layouts (CDNA5 ISA 7.12.2, 32-bit A 16x4 / B 4x16 / C,D 16x16):
//   A: lane&15 = M row, (lane>>4)*2 + vgpr = K
//   B: lane&15 = N col, (lane>>4)*2 + vgpr = K
//   D: lane&15 = N col, M = vgpr + 8*(lane>>4)
// ---------------------------------------------------------------------------
template <int D_IN, int D_HID, int D_OUT>
__global__ __launch_bounds__(256) void mlp_fused_kernel(
    const float* __restrict__ x,  const float* __restrict__ w1,
    const float* __restrict__ b1, const float* __restrict__ w2,
    const float* __restrict__ b2, float* __restrict__ out,
    const int* __restrict__ counts, const int* __restrict__ lists,
    int expert)
{
    extern __shared__ float h_lds[];            // [TILE_M][D_HID + LDS_PAD]
    __shared__ int tokS[TILE_M];

    constexpr int HSTRIDE = D_HID + LDS_PAD;

    const int cnt     = counts[expert];
    const int rowBase = blockIdx.x * TILE_M;
    if (rowBase >= cnt) return;
    const int nValid  = (cnt - rowBase < TILE_M) ? (cnt - rowBase) : TILE_M;

    // stage this tile's token ids (used for the scatter in pass 2)
    if (threadIdx.x < TILE_M) {
        int r = threadIdx.x;
        tokS[r] = lists[expert * TOKENS + rowBase + ((r < nValid) ? r : 0)];
    }

    const int lane = threadIdx.x & 31;
    const int wave = threadIdx.x >> 5;          // 0..7
    const int l15  = lane & 15;                 // A row / B col / D col
    const int kh   = lane >> 4;                 // K half (0,1)
    const int koff = 2 * kh;

    const int mClamped = (l15 < nValid) ? l15 : 0;
    const int tok      = lists[expert * TOKENS + rowBase + mClamped];
    const float* xrow  = x + (long long)tok * DIM;

    // ---------------- Pass 1: h = gelu(x @ w1^T + b1) ----------------
    constexpr int N_HID_TILES = D_HID >> 4;     // 32..256, multiple of 16
    for (int t0 = wave; t0 < N_HID_TILES; t0 += 16) {
        const int t1   = t0 + 8;
        const bool dual = (t1 < N_HID_TILES);
        const int j0 = t0 << 4;
        const int j1 = t1 << 4;
        const float* w1c0 = w1 + (long long)(j0 + l15) * DIM;
        const float* w1c1 = w1 + (long long)(j1 + l15) * DIM;
        v8f acc0 = {};
        v8f acc1 = {};
        if (dual) {
            #pragma unroll 4
            for (int k = 0; k < D_IN; k += 4) {
                const int kk = k + koff;
                v2f a  = *(const v2f*)(xrow + kk);     // shared A fragment
                v2f b0 = *(const v2f*)(w1c0 + kk);
                v2f b1v = *(const v2f*)(w1c1 + kk);
                acc0 = __builtin_amdgcn_wmma_f32_16x16x4_f32(
                    false, a, false, b0,  (short)0, acc0, false, false);
                acc1 = __builtin_amdgcn_wmma_f32_16x16x4_f32(
                    false, a, false, b1v, (short)0, acc1, false, false);
            }
        } else {
            #pragma unroll 4
            for (int k = 0; k < D_IN; k += 4) {
                const int kk = k + koff;
                v2f a  = *(const v2f*)(xrow + kk);
                v2f b0 = *(const v2f*)(w1c0 + kk);
                acc0 = __builtin_amdgcn_wmma_f32_16x16x4_f32(
                    false, a, false, b0, (short)0, acc0, false, false);
            }
        }
        const float bias0 = b1[j0 + l15];
        const float bias1 = dual ? b1[j1 + l15] : 0.0f;
        #pragma unroll
        for (int v = 0; v < 8; ++v) {
            const int row = v + 8 * kh;          // D-matrix M index
            h_lds[row * HSTRIDE + j0 + l15] = gelu_exact(acc0[v] + bias0);
            if (dual)
                h_lds[row * HSTRIDE + j1 + l15] = gelu_exact(acc1[v] + bias1);
        }
    }

    __syncthreads();

    // ---------------- Pass 2: y = h @ w2^T + b2 ----------------
    constexpr int N_OUT_TILES = D_OUT >> 4;     // 8..64
    for (int t0 = wave; t0 < N_OUT_TILES; t0 += 16) {
        const int t1   = t0 + 8;
        const bool dual = (t1 < N_OUT_TILES);
        const int o0 = t0 << 4;
        const int o1 = t1 << 4;
        const float* w2r0 = w2 + (long long)(o0 + l15) * HID;
        const float* w2r1 = w2 + (long long)(o1 + l15) * HID;
        const float* hrow = &h_lds[l15 * HSTRIDE];
        v8f acc0 = {};
        v8f acc1 = {};
        if (dual) {
            #pragma unroll 4
            for (int k = 0; k < D_HID; k += 4) {
                const int kk = k + koff;
                v2f a  = *(const v2f*)(hrow + kk);      // ds_load_b64, bank-clean
                v2f b0 = *(const v2f*)(w2r0 + kk);
                v2f b1v = *(const v2f*)(w2r1 + kk);
                acc0 = __builtin_amdgcn_wmma_f32_16x16x4_f32(
                    false, a, false, b0,  (short)0, acc0, false, false);
                acc1 = __builtin_amdgcn_wmma_f32_16x16x4_f32(
                    false, a, false, b1v, (short)0, acc1, false, false);
            }
        } else {
            #pragma unroll 4
            for (int k = 0; k < D_HID; k += 4) {
                const int kk = k + koff;
                v2f a  = *(const v2f*)(hrow + kk);
                v2f b0 = *(const v2f*)(w2r0 + kk);
                acc0 = __builtin_amdgcn_wmma_f32_16x16x4_f32(
                    false, a, false, b0, (short)0, acc0, false, false);
            }
        }
        const float bias0 = b2[o0 + l15];
        const float bias1 = dual ? b2[o1 + l15] : 0.0f;
        #pragma unroll
        for (int v = 0; v < 8; ++v) {
            const int row = v + 8 * kh;
            if (row < nValid) {
                long long base = (long long)tokS[row] * OUTD;
                out[base + o0 + l15] = acc0[v] + bias0;
                if (dual) out[base + o1 + l15] = acc1[v] + bias1;
            }
        }
    }
}

// ---------------------------------------------------------------------------
// Launch: zero -> route -> 4x fused expert GEMMs (all on `stream`)
// ---------------------------------------------------------------------------
extern "C" void kernel_launch(void* const* d_in, const int* in_sizes, int n_in,
                              void* d_out, int out_size, void* d_ws, size_t ws_size,
                              hipStream_t stream) {
    const float* x    = (const float*)d_in[0];
    const int*   mask = (const int*)  d_in[1];
    const float* w1   = (const float*)d_in[2];
    const float* b1   = (const float*)d_in[3];
    const float* w2   = (const float*)d_in[4];
    const float* b2   = (const float*)d_in[5];
    float*       out  = (float*)d_out;

    int* counts = (int*)d_ws;                       // 4 ints
    int* lists  = (int*)((char*)d_ws + 256);        // 4 * TOKENS ints (256 KB)

    // 1) zero output (16384*1024 f32 = 4M float4) + counters
    {
        const int n4 = TOKENS * OUTD / 4;
        zero_out_kernel<<<(n4 + 255) / 256, 256, 0, stream>>>(out, n4, counts);
    }
    // 2) route tokens to experts
    route_kernel<<<(TOKENS + 255) / 256, 256, 0, stream>>>(mask, counts, lists);

    // 3) per-expert fused MLP; worst-case tile grid, blocks early-exit on count
    const int nTiles = TOKENS / TILE_M;             // 1024
    const size_t sh0 = (size_t)TILE_M * ( 512 + LDS_PAD) * sizeof(float);
    const size_t sh1 = (size_t)TILE_M * (1024 + LDS_PAD) * sizeof(float);
    const size_t sh2 = (size_t)TILE_M * (2048 + LDS_PAD) * sizeof(float);
    const size_t sh3 = (size_t)TILE_M * (4096 + LDS_PAD) * sizeof(float);
    mlp_fused_kernel< 128,  512,  128><<<nTiles, 256, sh0, stream>>>(
        x, w1, b1, w2, b2, out, counts, lists, 0);
    mlp_fused_kernel< 256, 1024,  256><<<nTiles, 256, sh1, stream>>>(
        x, w1, b1, w2, b2, out, counts, lists, 1);
    mlp_fused_kernel< 512, 2048,  512><<<nTiles, 256, sh2, stream>>>(
        x, w1, b1, w2, b2, out, counts, lists, 2);
    mlp_fused_kernel<1024, 4096, 1024><<<nTiles, 256, sh3, stream>>>(
        x, w1, b1, w2, b2, out, counts, lists, 3);
}